// LayerTree_67430986547632
// MI455X (gfx1250) — compile-verified
//
#include <hip/hip_runtime.h>

typedef __attribute__((ext_vector_type(2))) float v2f;
typedef __attribute__((ext_vector_type(8))) float v8f;
typedef __attribute__((ext_vector_type(4))) unsigned int v4u;
typedef __attribute__((ext_vector_type(8))) int v8i;
typedef __attribute__((ext_vector_type(4))) int v4i;

#define BM 128
#define BN 128
#define BK 32
#define LDA_S 36   // sA row stride in floats = 32 + TDM pad (4 dw after every 32 dw)
#define LDB_S 132  // sB row stride in floats = 128 + TDM pad (4 dw after every 128 dw)

// Issue one TDM 2D tile load: tile0 elements contiguous (dim0), tile1 lines with
// stride stride0 (in elements), fp32, LDS padding padAmt dwords every padIvl code.
__device__ __forceinline__ void tdm_load_2d(
    unsigned ldsByteAddr, const float* gbase,
    unsigned tile0, unsigned tile1, unsigned long long stride0,
    unsigned padIvlCode, unsigned padAmtCode)
{
  unsigned long long ga = (unsigned long long)(uintptr_t)gbase;

  v4u g0;
  g0[0] = 1u;                                              // count=1, user mode
  g0[1] = ldsByteAddr;                                     // lds_addr [63:32]
  g0[2] = (unsigned)(ga & 0xFFFFFFFFu);                    // global_addr lo
  g0[3] = (unsigned)((ga >> 32) & 0x01FFFFFFu) | (2u << 30); // ga[56:32] | type=2

  // group1: data_size=2 (4B), pad_enable, pad interval/amount, dims & stride.
  unsigned cfg = (2u << 16) | (1u << 20) | (padIvlCode << 22) | (padAmtCode << 25);
  unsigned long long td0 = tile0, td1 = tile1;             // tensor dims = tile dims
  unsigned long long q0 = (unsigned long long)cfg | ((td0 & 0xFFFFull) << 48);
  unsigned long long q1 = (td0 >> 16) | (td1 << 16) | ((unsigned long long)tile0 << 48);
  unsigned long long q2 = (unsigned long long)tile1 | ((stride0 & 0xFFFFFFFFull) << 32);
  unsigned long long q3 = (stride0 >> 32) & 0xFFFFull;     // dim1_stride=0 (2D tile)

  v8i g1;
  g1[0] = (int)(unsigned)q0; g1[1] = (int)(unsigned)(q0 >> 32);
  g1[2] = (int)(unsigned)q1; g1[3] = (int)(unsigned)(q1 >> 32);
  g1[4] = (int)(unsigned)q2; g1[5] = (int)(unsigned)(q2 >> 32);
  g1[6] = (int)(unsigned)q3; g1[7] = (int)(unsigned)(q3 >> 32);

  v4i g2 = {0, 0, 0, 0};
  v4i g3 = {0, 0, 0, 0};
#if defined(__clang_major__) && (__clang_major__ >= 23)
  v8i g4 = {0, 0, 0, 0, 0, 0, 0, 0};
  __builtin_amdgcn_tensor_load_to_lds(g0, g1, g2, g3, g4, 0);
#else
  __builtin_amdgcn_tensor_load_to_lds(g0, g1, g2, g3, 0);
#endif
}

// flat = A(B x K) @ Wcat(K x N) + bias via V_WMMA_F32_16X16X4_F32.
// A/B tiles staged to LDS by the Tensor Data Mover, double buffered on TENSORcnt.
__global__ __launch_bounds__(256) void gemm_bias_wmma(
    const float* __restrict__ A, int lda,
    const float* __restrict__ W,
    const float* __restrict__ bias,
    float* __restrict__ Cout,
    int K, int N, int dout)
{
  __shared__ __align__(16) float sA[2][BM * LDA_S];
  __shared__ __align__(16) float sB[2][BK * LDB_S];

  const int tid      = threadIdx.x;
  const int lane     = tid & 31;
  const int wv       = tid >> 5;      // 0..7 waves
  const int waveM    = wv >> 1;       // 0..3
  const int waveN    = wv & 1;        // 0..1
  const int laneRow  = lane & 15;
  const int laneHalf = lane >> 4;

  const int tileN   = blockIdx.x;
  const int tileM   = blockIdx.y;
  const int rowBase = waveM * 32;
  const int colBase = waveN * 64;

  const int gcol0  = tileN * BN;
  const int nblk   = gcol0 / dout;
  const int cstart = gcol0 % dout;
  const float* Wt = W + (size_t)nblk * K * dout + cstart;   // rows K, stride dout
  const float* At = A + (size_t)(tileM * BM) * lda;

  const unsigned ldsA0 = (unsigned)(uintptr_t)(void*)&sA[0][0];
  const unsigned ldsA1 = (unsigned)(uintptr_t)(void*)&sA[1][0];
  const unsigned ldsB0 = (unsigned)(uintptr_t)(void*)&sB[0][0];
  const unsigned ldsB1 = (unsigned)(uintptr_t)(void*)&sB[1][0];

  // Accumulators initialized with the bias (bias depends only on the column)
  v8f acc[2][4];
#pragma unroll
  for (int ni = 0; ni < 4; ++ni) {
    float bv = bias[gcol0 + colBase + ni * 16 + laneRow];
#pragma unroll
    for (int mi = 0; mi < 2; ++mi) {
      v8f t;
#pragma unroll
      for (int v = 0; v < 8; ++v) t[v] = bv;
      acc[mi][ni] = t;
    }
  }

  const int nkb = K / BK;

  // Prologue: wave0 kicks TDM loads for kb=0 into buffer 0.
  if (wv == 0) {
    tdm_load_2d(ldsA0, At, BK, BM, (unsigned long long)lda, 4u /*32dw*/, 3u /*4dw*/);
    tdm_load_2d(ldsB0, Wt, BN, BK, (unsigned long long)dout, 6u /*128dw*/, 3u /*4dw*/);
  }

  for (int kb = 0; kb < nkb; ++kb) {
    if (wv == 0) {
      if (kb + 1 < nkb) {
        // Prefetch next tile into the other buffer, then wait until only those
        // two TDM ops remain outstanding (==> kb's tiles have landed in LDS).
        const int nb = (kb + 1) & 1;
        tdm_load_2d(nb ? ldsA1 : ldsA0, At + (size_t)(kb + 1) * BK,
                    BK, BM, (unsigned long long)lda, 4u, 3u);
        tdm_load_2d(nb ? ldsB1 : ldsB0, Wt + (size_t)(kb + 1) * BK * dout,
                    BN, BK, (unsigned long long)dout, 6u, 3u);
        __builtin_amdgcn_s_wait_tensorcnt(2);
      } else {
        __builtin_amdgcn_s_wait_tensorcnt(0);
      }
    }
    __syncthreads();                       // publish kb's LDS tiles to all waves

    const float* cA = sA[kb & 1];
    const float* cB = sB[kb & 1];

#pragma unroll
    for (int kk = 0; kk < BK; kk += 4) {
      v2f af[2];
#pragma unroll
      for (int mi = 0; mi < 2; ++mi) {
        int M = rowBase + mi * 16 + laneRow;
        af[mi] = *(const v2f*)&cA[M * LDA_S + kk + laneHalf * 2];
      }
#pragma unroll
      for (int ni = 0; ni < 4; ++ni) {
        int Ncol = colBase + ni * 16 + laneRow;
        int krow = kk + laneHalf * 2;
        v2f bf;
        bf.x = cB[krow * LDB_S + Ncol];
        bf.y = cB[(krow + 1) * LDB_S + Ncol];
#pragma unroll
        for (int mi = 0; mi < 2; ++mi) {
          acc[mi][ni] = __builtin_amdgcn_wmma_f32_16x16x4_f32(
              false, af[mi], false, bf, (short)0, acc[mi][ni], false, false);
        }
      }
    }

    __syncthreads();   // all waves done with buf[kb&1] before TDM reuses it (kb+2)
  }

  // Epilogue: C/D layout -> VGPR v holds (M = v + 8*laneHalf, N = laneRow)
  const int gr0 = tileM * BM + rowBase;
#pragma unroll
  for (int mi = 0; mi < 2; ++mi) {
#pragma unroll
    for (int ni = 0; ni < 4; ++ni) {
      int col = gcol0 + colBase + ni * 16 + laneRow;
#pragma unroll
      for (int v = 0; v < 8; ++v) {
        int row = gr0 + mi * 16 + laneHalf * 8 + v;
        Cout[(size_t)row * N + col] = acc[mi][ni][v];
      }
    }
  }
}

// probs[b, i] = flat[b,:] @ Wc[:, i] + bc[i]; writes ws copy (stride 8) + output slot.
template <int NODES>
__global__ __launch_bounds__(256) void classifier_kernel(
    const float* __restrict__ flat, const float* __restrict__ Wc,
    const float* __restrict__ bc, float* __restrict__ probs8,
    float* __restrict__ outP, int N)
{
  __shared__ float red[8][NODES];
  const int b = blockIdx.x;
  const int tid = threadIdx.x;
  float acc[NODES];
#pragma unroll
  for (int i = 0; i < NODES; ++i) acc[i] = 0.f;
  const float* fr = flat + (size_t)b * N;
  for (int j = tid; j < N; j += 256) {
    float x = fr[j];
#pragma unroll
    for (int i = 0; i < NODES; ++i) acc[i] += x * Wc[(size_t)j * NODES + i];
  }
#pragma unroll
  for (int off = 16; off > 0; off >>= 1)
#pragma unroll
    for (int i = 0; i < NODES; ++i) acc[i] += __shfl_down(acc[i], off, 32);
  const int lane = tid & 31, w = tid >> 5;
  if (lane == 0)
#pragma unroll
    for (int i = 0; i < NODES; ++i) red[w][i] = acc[i];
  __syncthreads();
  if (tid < NODES) {
    float s = bc[tid];
#pragma unroll
    for (int q = 0; q < 8; ++q) s += red[q][tid];
    probs8[b * 8 + tid] = s;
    outP[(size_t)b * 12306 + tid] = s;
  }
}

// gated = flat * probs[node]; writes into d_out emb region (stride 12306), 2 elems/thread.
__global__ __launch_bounds__(256) void gate_kernel(
    const float* __restrict__ flat, const float* __restrict__ probs8,
    float* __restrict__ outEmb, int N, int nShift, int dShift)
{
  unsigned idx2 = blockIdx.x * 256u + threadIdx.x;          // pair index
  unsigned b  = idx2 >> (nShift - 1);
  unsigned j  = (idx2 & ((N >> 1) - 1)) << 1;               // even column
  v2f f = *(const v2f*)(flat + (size_t)b * N + j);
  float p = probs8[b * 8 + (j >> dShift)];
  v2f g; g.x = f.x * p; g.y = f.y * p;
  *(v2f*)(outEmb + (size_t)b * 12306 + j) = g;              // 12306 even -> 8B aligned
}

extern "C" void kernel_launch(void* const* d_in, const int* in_sizes, int n_in,
                              void* d_out, int out_size, void* d_ws, size_t ws_size,
                              hipStream_t stream) {
  (void)in_sizes; (void)n_in; (void)out_size; (void)ws_size;
  const float* x = (const float*)d_in[0];
  float* out = (float*)d_out;
  float* flat = (float*)d_ws;                       // 8192*4096 floats (128 MB)
  float* probs8 = flat + (size_t)8192 * 4096;       // 8192*8 floats

  struct L { int n, din, dout, embOff, probOff, nShift, dShift; };
  const L Ls[4] = {
    {2, 4096, 1024,    0, 12288, 11, 10},
    {4, 2048,  512, 2048, 12290, 11,  9},
    {8, 2048,  512, 4096, 12294, 12,  9},
    {4, 4096, 1024, 8192, 12302, 12, 10},
  };
  const int Bdim = 8192;

  const float* Aptr = x;
  int lda = 4096;
  for (int l = 0; l < 4; ++l) {
    const float* W  = (const float*)d_in[1 + 4 * l + 0];
    const float* bb = (const float*)d_in[1 + 4 * l + 1];
    const float* Wc = (const float*)d_in[1 + 4 * l + 2];
    const float* bc = (const float*)d_in[1 + 4 * l + 3];
    const int N = Ls[l].n * Ls[l].dout;

    dim3 g(N / BN, Bdim / BM);
    gemm_bias_wmma<<<g, 256, 0, stream>>>(Aptr, lda, W, bb, flat, Ls[l].din, N, Ls[l].dout);

    switch (Ls[l].n) {
      case 2: classifier_kernel<2><<<Bdim, 256, 0, stream>>>(flat, Wc, bc, probs8, out + Ls[l].probOff, N); break;
      case 4: classifier_kernel<4><<<Bdim, 256, 0, stream>>>(flat, Wc, bc, probs8, out + Ls[l].probOff, N); break;
      case 8: classifier_kernel<8><<<Bdim, 256, 0, stream>>>(flat, Wc, bc, probs8, out + Ls[l].probOff, N); break;
    }

    const unsigned tot2 = (unsigned)Bdim * (unsigned)N / 2u;
    gate_kernel<<<dim3(tot2 / 256u), 256, 0, stream>>>(flat, probs8, out + Ls[l].embOff, N, Ls[l].nShift, Ls[l].dShift);

    Aptr = out + Ls[l].embOff;   // gated embeddings are next level's input
    lda = 12306;
  }
}